// AngularPenaltySMLoss_72507637891691
// MI455X (gfx1250) — compile-verified
//
#include <hip/hip_runtime.h>

typedef float v2f __attribute__((ext_vector_type(2)));
typedef float v8f __attribute__((ext_vector_type(8)));

#define S_SCALE   30.0f
#define EPS_CLIP  1e-7f
// cos(0.5), sin(0.5) precomputed (M = 0.5 fixed)
#define COS_M     0.87758256189037271612f
#define SIN_M     0.47942553860420300027f

#define NBLOCKS   1024
#define NTHREADS  256
#define WAVES_PER_BLOCK (NTHREADS / 32)

// Kernel 1: streaming pass. Each wave handles 32 rows/iter:
//  - per-lane: load x, norm via v_rsq_f32, soft-penalty, normalize
//  - two V_WMMA_F32_16X16X4_F32 compute wf = xn @ W.T for rows 0..15 / 16..31
//  - D tile spilled to per-wave LDS so lane i owns row i's 16 (10 valid) logits
//  - per-lane ArcFace margin + log-sum-exp epilogue, double accumulation
__global__ __launch_bounds__(NTHREADS) void angular_partial_kernel(
    const float* __restrict__ x,       // [N,2]
    const int*   __restrict__ labels,  // [N]
    const float* __restrict__ weight,  // [10,2]
    double*      __restrict__ partial, // [gridDim.x]
    int N)
{
    __shared__ float  lds_wf[WAVES_PER_BLOCK][512]; // 32 rows x 16 cols per wave
    __shared__ double lds_red[WAVES_PER_BLOCK];

    const int lane = threadIdx.x & 31;
    const int wid  = threadIdx.x >> 5;
    float* wf = lds_wf[wid];

    // B operand (4x16 f32): lane l<16 holds {B[0][l], B[1][l]} = {W[l][0], W[l][1]}
    // for class l (zero for l>=10); lanes 16..31 hold K=2,3 rows = zero.
    int widx = (lane < 10) ? lane : 0;
    float w0 = weight[2 * widx + 0];
    float w1 = weight[2 * widx + 1];
    v2f bmat;
    bmat.x = (lane < 10) ? w0 : 0.0f;
    bmat.y = (lane < 10) ? w1 : 0.0f;

    const int gwave  = blockIdx.x * WAVES_PER_BLOCK + wid;
    const int nwaves = gridDim.x * WAVES_PER_BLOCK;

    double acc = 0.0;

    for (int base = gwave * 32; base < N; base += nwaves * 32) {
        const int  row = base + lane;
        const bool inb = row < N;
        const int  r   = inb ? row : (N - 1);      // clamped, branchless load

        float2 xv = ((const float2*)x)[r];          // global_load_b64
        int lab = labels[r];
        lab = lab < 0 ? 0 : (lab > 9 ? 9 : lab);

        // norm + reciprocal in one v_rsq_f32: rn = 1/sqrt(n2), n = n2*rn = sqrt(n2)
        // clamping n2 at 1e-24 reproduces max(norm, 1e-12)
        float n2  = fmaf(xv.x, xv.x, xv.y * xv.y);
        float rn  = __builtin_amdgcn_rsqf(fmaxf(n2, 1e-24f));
        float n   = n2 * rn;
        float v   = n - 1.0f;
        float soft = (v < 0.5f ? (0.5f - v) : 0.0f) + (v > 1.0f ? (v - 1.0f) : 0.0f);
        float xn0 = xv.x * rn;
        float xn1 = xv.y * rn;

        // A operands (16x4 f32): lanes 0..15 = {A[m][0], A[m][1]}, lanes 16..31 = K=2,3 = 0
        v2f a1, a2;
        a1.x = (lane < 16) ? xn0 : 0.0f;
        a1.y = (lane < 16) ? xn1 : 0.0f;
        float ox0 = __shfl_xor(xn0, 16, 32);        // rows 16..31 into lanes 0..15
        float ox1 = __shfl_xor(xn1, 16, 32);
        a2.x = (lane < 16) ? ox0 : 0.0f;
        a2.y = (lane < 16) ? ox1 : 0.0f;

        v8f cz = {0.f, 0.f, 0.f, 0.f, 0.f, 0.f, 0.f, 0.f};
        v8f d1 = __builtin_amdgcn_wmma_f32_16x16x4_f32(false, a1, false, bmat,
                                                       (short)0, cz, false, false);
        v8f d2 = __builtin_amdgcn_wmma_f32_16x16x4_f32(false, a2, false, bmat,
                                                       (short)0, cz, false, false);

        // D layout: lane<16 -> D[rr][lane] in d[rr]; lane>=16 -> D[rr+8][lane-16]
        const int col   = lane & 15;
        const int rbase = (lane < 16) ? 0 : 8;
        #pragma unroll
        for (int rr = 0; rr < 8; ++rr) {
            wf[(rbase + rr) * 16 + col]        = d1[rr];
            wf[(16 + rbase + rr) * 16 + col]   = d2[rr];
        }
        asm volatile("" ::: "memory");  // per-wave LDS ops are HW in-order; block compiler reorder

        // Lane i owns row i: 64B-aligned row of 16 floats (10 valid)
        const float* myrow = &wf[lane * 16];
        float4 wa = *(const float4*)(myrow + 0);
        float4 wb = *(const float4*)(myrow + 4);
        float2 wc = *(const float2*)(myrow + 8);
        float  tgt = myrow[lab];                    // label gather = indexed ds_load_b32

        float se = __expf(S_SCALE * wa.x) + __expf(S_SCALE * wa.y) +
                   __expf(S_SCALE * wa.z) + __expf(S_SCALE * wa.w) +
                   __expf(S_SCALE * wb.x) + __expf(S_SCALE * wb.y) +
                   __expf(S_SCALE * wb.z) + __expf(S_SCALE * wb.w) +
                   __expf(S_SCALE * wc.x) + __expf(S_SCALE * wc.y);

        // cos(acos(t)+M) = t*cosM - sqrt(1-t^2)*sinM  (no acos/cos needed)
        float t     = fminf(fmaxf(tgt, -1.0f + EPS_CLIP), 1.0f - EPS_CLIP);
        float sq    = __builtin_amdgcn_sqrtf(fmaxf(1.0f - t * t, 0.0f)); // raw v_sqrt_f32
        float num   = S_SCALE * (t * COS_M - sq * SIN_M);
        float excl  = se - __expf(S_SCALE * tgt);
        float denom = __expf(num) + excl;
        float L     = num - __logf(denom);

        float contrib = -L + 0.5f * soft;           // LBDA*soft/2 with LBDA=1
        acc += inb ? (double)contrib : 0.0;
    }

    // wave32 tree reduction, then cross-wave via LDS
    #pragma unroll
    for (int off = 16; off > 0; off >>= 1)
        acc += __shfl_down(acc, off, 32);
    if (lane == 0) lds_red[wid] = acc;
    __syncthreads();
    if (threadIdx.x == 0) {
        double s = 0.0;
        #pragma unroll
        for (int w = 0; w < WAVES_PER_BLOCK; ++w) s += lds_red[w];
        partial[blockIdx.x] = s;
    }
}

// Kernel 2: deterministic single-block finish: out = sum(partials) / N
__global__ __launch_bounds__(NTHREADS) void angular_reduce_kernel(
    const double* __restrict__ partial, int nparts,
    float* __restrict__ out, double invN)
{
    __shared__ double s[WAVES_PER_BLOCK];
    double a = 0.0;
    for (int i = threadIdx.x; i < nparts; i += NTHREADS) a += partial[i];
    #pragma unroll
    for (int off = 16; off > 0; off >>= 1)
        a += __shfl_down(a, off, 32);
    const int lane = threadIdx.x & 31;
    const int wid  = threadIdx.x >> 5;
    if (lane == 0) s[wid] = a;
    __syncthreads();
    if (threadIdx.x == 0) {
        double t = 0.0;
        #pragma unroll
        for (int w = 0; w < WAVES_PER_BLOCK; ++w) t += s[w];
        out[0] = (float)(t * invN);
    }
}

extern "C" void kernel_launch(void* const* d_in, const int* in_sizes, int n_in,
                              void* d_out, int out_size, void* d_ws, size_t ws_size,
                              hipStream_t stream)
{
    (void)n_in; (void)out_size; (void)ws_size;
    const float* x      = (const float*)d_in[0];
    const int*   labels = (const int*)d_in[1];
    const float* weight = (const float*)d_in[2];
    float*  out     = (float*)d_out;
    double* partial = (double*)d_ws;   // NBLOCKS doubles = 8 KB scratch

    const int N = in_sizes[1];         // labels count == N rows

    angular_partial_kernel<<<NBLOCKS, NTHREADS, 0, stream>>>(x, labels, weight, partial, N);
    angular_reduce_kernel<<<1, NTHREADS, 0, stream>>>(partial, NBLOCKS, out, 1.0 / (double)N);
}